// DetectionLoss_85770496901338
// MI455X (gfx1250) — compile-verified
//
#include <hip/hip_runtime.h>
#include <cstdint>
#include <cstddef>

// ---------------- problem constants (from reference) ----------------
#define B_TOT   16384
#define NCLS    24
#define GRID7   7
#define GG      49            // 7*7
#define CHANS   34            // 24 classes + 2 conf + 8 box
#define IMGF    448.0f
#define CELLF   64.0f         // 448/7
#define NB3     512           // blocks for the streaming object-loss pass

typedef __attribute__((ext_vector_type(2))) float v2f;
typedef __attribute__((ext_vector_type(8))) float v8f;

// -------------------------------------------------------------------
// Wave32 + block sum using the matrix pipe:
// A (16x4 f32): lane<16 holds rows 0..15 at K0 (K1=0); lane>=16 holds K2 (K3=0).
// B = ones(4x16)  =>  D[m,n] = sum_k A[m,k]  (exact f32 row-sums).
// Summing the 8 D VGPRs gives lanes 0..15: sum(acc0..7 + acc16..23),
// lanes 16..31: sum(acc8..15 + acc24..31); xor-shuffle by 16 completes the wave.
// -------------------------------------------------------------------
__device__ __forceinline__ float block_sum_wmma(float v) {
  v2f a;     a[0] = v;    a[1] = 0.0f;
  v2f bones; bones[0] = 1.0f; bones[1] = 1.0f;
  v8f c = {};
  c = __builtin_amdgcn_wmma_f32_16x16x4_f32(false, a, false, bones,
                                            (short)0, c, false, false);
  float s = ((c[0] + c[1]) + (c[2] + c[3])) + ((c[4] + c[5]) + (c[6] + c[7]));
  s += __shfl_xor(s, 16, 32);                 // full wave32 sum on every lane

  __shared__ float wsum[8];                   // up to 8 waves / 256-thread block
  const int lane = threadIdx.x & 31;
  const int wid  = threadIdx.x >> 5;
  if (lane == 0) wsum[wid] = s;
  __syncthreads();
  float tot = 0.0f;
  if (threadIdx.x == 0) {
    const int nw = (blockDim.x + 31) >> 5;
    for (int w = 0; w < nw; ++w) tot += wsum[w];
  }
  return tot;                                  // valid on thread 0
}

// ---------------- pass 0: re-init the scatter winner grid ----------------
__global__ void init_kernel(int* __restrict__ winner) {
  if (threadIdx.x < GG) winner[threadIdx.x] = -1;
}

// ---------------- pass 1: per-sample gather (coord + class loss) ----------------
__global__ __launch_bounds__(256) void per_sample_kernel(
    const float* __restrict__ outp, const float* __restrict__ tcoords,
    const int* __restrict__ label, float* __restrict__ conf_ws,
    int* __restrict__ winner, float* __restrict__ p1) {
  const int b = blockIdx.x * blockDim.x + threadIdx.x;
  float local = 0.0f;
  if (b < B_TOT) {
    const float x0 = tcoords[4 * b + 0];
    const float y0 = tcoords[4 * b + 1];
    const float tw = tcoords[4 * b + 2];
    const float th = tcoords[4 * b + 3];
    const float txc = fminf(x0 + 0.5f * tw, IMGF - 1.0f);
    const float tyc = fminf(y0 + 0.5f * th, IMGF - 1.0f);
    const int cx = (int)floorf(txc * (1.0f / CELLF));
    const int cy = (int)floorf(tyc * (1.0f / CELLF));
    const float tx = txc - (float)cx * CELLF;
    const float ty = tyc - (float)cy * CELLF;
    const int cell = cx * GRID7 + cy;
    const float* samp = outp + (size_t)b * (CHANS * GG) + cell;

    // pred channels 26..33 at the responsible cell: x0 x1 y0 y1 w0 w1 h0 h1
    float px[2], py[2], pw[2], ph[2];
#pragma unroll
    for (int j = 0; j < 2; ++j) {
      px[j] = samp[(26 + j) * GG];
      py[j] = samp[(28 + j) * GG];
      pw[j] = samp[(30 + j) * GG];
      ph[j] = samp[(32 + j) * GG];
    }
    const float tx1 = tx - 0.5f * tw, tx2 = tx + 0.5f * tw;
    const float ty1 = ty - 0.5f * th, ty2 = ty + 0.5f * th;
    const float area_t = tw * th;
    float iou[2];
#pragma unroll
    for (int j = 0; j < 2; ++j) {
      const float px1 = px[j] - 0.5f * pw[j], px2 = px[j] + 0.5f * pw[j];
      const float py1 = py[j] - 0.5f * ph[j], py2 = py[j] + 0.5f * ph[j];
      const float dx = fmaxf(0.0f, fminf(tx2, px2) - fminf(IMGF, fmaxf(tx1, px1)));
      const float dy = fmaxf(0.0f, fminf(ty2, py2) - fminf(IMGF, fmaxf(ty1, py1)));
      const float inter = dx * dy;
      const float uni = area_t + pw[j] * ph[j] - inter;
      const float v = inter / uni;
      iou[j] = (v < 1e-6f) ? 0.0f : v;
    }
    const int resp = (iou[1] > iou[0]) ? 1 : 0;   // argmax, first-wins on tie
    conf_ws[b] = fmaxf(iou[0], iou[1]);
    atomicMax(&winner[cell], b);                   // last batch index wins the tc scatter

    // coord loss (only the responsible (cell, box) survives I_obj masking)
    {
      const float pwc = fminf(fmaxf(pw[resp], 0.0f), IMGF);
      const float phc = fminf(fmaxf(ph[resp], 0.0f), IMGF);
      const float psw = (pwc > 0.0f) ? sqrtf(pwc) : 0.0f;   // safe_sqrt
      const float psh = (phc > 0.0f) ? sqrtf(phc) : 0.0f;
      const float d0 = tx - px[resp];
      const float d1 = ty - py[resp];
      const float d2 = sqrtf(tw) - psw;   // tw, th > 0 by construction
      const float d3 = sqrtf(th) - psh;
      local = d0 * d0 + d1 * d1 + d2 * d2 + d3 * d3;
    }

    // class loss: only counted where I_obj[...,0]==1, i.e. resp==0.
    // sum_c (softmax_c - onehot_c)^2 = sum softmax^2 - 2*softmax[label] + 1
    if (resp == 0) {
      float cls[NCLS];
      float mx = -3.4e38f;
#pragma unroll
      for (int k = 0; k < NCLS; ++k) { cls[k] = samp[k * GG]; mx = fmaxf(mx, cls[k]); }
      float se = 0.0f;
#pragma unroll
      for (int k = 0; k < NCLS; ++k) { cls[k] = expf(cls[k] - mx); se += cls[k]; }
      const float inv = 1.0f / se;
      float s2 = 0.0f;
#pragma unroll
      for (int k = 0; k < NCLS; ++k) { const float sm = cls[k] * inv; s2 = fmaf(sm, sm, s2); }
      const int lbl = label[b] - 1;
      local += s2 - 2.0f * (cls[lbl] * inv) + 1.0f;
    }
  }
  const float tot = block_sum_wmma(local);   // EXEC fully reconverged here
  if (threadIdx.x == 0) p1[blockIdx.x] = tot;
}

// ---------------- pass 2: resolve tc grid from scatter winners ----------------
__global__ void tc_kernel(const int* __restrict__ winner,
                          const float* __restrict__ conf_ws,
                          float* __restrict__ tcg) {
  const int i = threadIdx.x;
  if (i < GG) {
    const int w = winner[i];
    tcg[i] = (w >= 0) ? conf_ws[w] : 0.0f;
  }
}

// ---------------- pass 3: streaming object loss ----------------
// NOOBJ_SCALE==1 collapses obj+noobj terms to sum over ALL (b, box, cell) of
// (tc[cell] - region_conf)^2. region_conf element for flat i = b*98 + r is at
// b*34*49 + 24*49 + r  (r in [0,98), cell = r % 49).
__global__ __launch_bounds__(256) void object_loss_kernel(
    const float* __restrict__ outp, const float* __restrict__ tcg,
    float* __restrict__ p3) {
  __shared__ float tcs[GG];
  if (threadIdx.x < GG) tcs[threadIdx.x] = tcg[threadIdx.x];
  __syncthreads();
  const long N3 = (long)B_TOT * 98;
  const long stride = (long)gridDim.x * blockDim.x;
  float acc = 0.0f;
  for (long i = (long)blockIdx.x * blockDim.x + threadIdx.x; i < N3; i += stride) {
    const long bi = i / 98;
    const int r = (int)(i - bi * 98);
    const int g = r % GG;
    const size_t addr = (size_t)bi * (CHANS * GG) + 24 * GG + r;
    // cover HBM latency for the next grid-stride iteration
    const long in = i + stride;
    if (in < N3) {
      const long b2 = in / 98;
      const int r2 = (int)(in - b2 * 98);
      __builtin_prefetch(outp + ((size_t)b2 * (CHANS * GG) + 24 * GG + r2), 0, 1);
    }
    const float e = tcs[g] - outp[addr];
    acc = fmaf(e, e, acc);
  }
  const float tot = block_sum_wmma(acc);
  if (threadIdx.x == 0) p3[blockIdx.x] = tot;
}

// ---------------- pass 4: final reduction to scalar ----------------
__global__ __launch_bounds__(256) void final_reduce_kernel(
    const float* __restrict__ p1, int n1,
    const float* __restrict__ p3, int n3, float* __restrict__ out) {
  float acc = 0.0f;
  for (int i = threadIdx.x; i < n1; i += blockDim.x) acc += p1[i];
  for (int i = threadIdx.x; i < n3; i += blockDim.x) acc += p3[i];
  const float tot = block_sum_wmma(acc);
  if (threadIdx.x == 0) out[0] = tot;
}

// -------------------------------------------------------------------
extern "C" void kernel_launch(void* const* d_in, const int* in_sizes, int n_in,
                              void* d_out, int out_size, void* d_ws, size_t ws_size,
                              hipStream_t stream) {
  const float* outp    = (const float*)d_in[0];   // (B, 34, 7, 7) f32
  const float* tcoords = (const float*)d_in[1];   // (B, 4) f32
  const int*   label   = (const int*)d_in[2];     // (B,) i32
  float* out = (float*)d_out;

  // workspace layout (all fully rewritten every call)
  float* conf_ws = (float*)d_ws;                  // [B_TOT] per-sample conf_iou
  int*   winner  = (int*)(conf_ws + B_TOT);       // [64]  (49 used)
  float* tcg     = (float*)(winner + 64);         // [64]  (49 used)
  float* p1      = tcg + 64;                      // [64]  block partials, pass 1
  float* p3      = p1 + 64;                       // [NB3] block partials, pass 3

  const int nb1 = B_TOT / 256;                    // 64 full blocks

  init_kernel<<<1, 64, 0, stream>>>(winner);
  per_sample_kernel<<<nb1, 256, 0, stream>>>(outp, tcoords, label,
                                             conf_ws, winner, p1);
  tc_kernel<<<1, 64, 0, stream>>>(winner, conf_ws, tcg);
  object_loss_kernel<<<NB3, 256, 0, stream>>>(outp, tcg, p3);
  final_reduce_kernel<<<1, 256, 0, stream>>>(p1, nb1, p3, NB3, out);
}